// TIDE_62672162783962
// MI455X (gfx1250) — compile-verified
//
#include <hip/hip_runtime.h>
#include <stdint.h>

// ---------------------------------------------------------------------------
// out = x  (vectorized float4 copy; out must be fully initialized since some
// nodes receive no edges)
// ---------------------------------------------------------------------------
__global__ void tide_init_out(const float* __restrict__ x,
                              float* __restrict__ out, int n4) {
    int i = blockIdx.x * blockDim.x + threadIdx.x;
    if (i < n4) {
        const float4* xv = (const float4*)x;
        float4* ov = (float4*)out;
        ov[i] = xv[i];
    }
}

// ---------------------------------------------------------------------------
// Edge scatter kernel, double-buffered.
//   - Each block handles 2 tiles of 512 edges (256 threads = 8 waves).
//   - Edge metadata (row/col low dwords interleaved as int2, plus weight) is
//     staged into LDS with gfx1250 async global->LDS copies (ASYNCcnt path);
//     tile t+1's staging overlaps with tile t's processing.
//   - Per edge: one wave; 32 lanes cover the 128 features as float4 each:
//     coalesced 512B b128 gather of x[col], then 4 no-return hardware
//     global_atomic_add_f32 into out[row].
// ---------------------------------------------------------------------------
#define TILE     512
#define THREADS  256
#define TPB      2      // tiles per block (double buffer)

__device__ __forceinline__ void stage_tile(const int* __restrict__ ei32,
                                           const float* __restrict__ ew,
                                           int E, int tileBase, int tid,
                                           uint32_t lds_rc, uint32_t lds_w) {
    #pragma unroll
    for (int j = 0; j < TILE / THREADS; ++j) {
        const int li = tid + j * THREADS;
        const int e  = tileBase + li;
        if (e < E) {
            const int*   pr = ei32 + 2 * (size_t)e;               // row lo32
            const int*   pc = ei32 + 2 * ((size_t)E + (size_t)e); // col lo32
            const float* pw = ew + e;
            const uint32_t drc = lds_rc + 8u * (uint32_t)li;      // int2 slot
            const uint32_t dcc = drc + 4u;
            const uint32_t dw  = lds_w + 4u * (uint32_t)li;
            asm volatile("global_load_async_to_lds_b32 %0, %1, off"
                         :: "v"(drc), "v"(pr) : "memory");
            asm volatile("global_load_async_to_lds_b32 %0, %1, off"
                         :: "v"(dcc), "v"(pc) : "memory");
            asm volatile("global_load_async_to_lds_b32 %0, %1, off"
                         :: "v"(dw), "v"(pw) : "memory");
        }
    }
}

__global__ void __launch_bounds__(THREADS)
tide_edge_scatter(const float* __restrict__ x,
                  const long long* __restrict__ edge_index, // [2, E] int64
                  const float* __restrict__ edge_weight,    // [E]
                  const float* __restrict__ theta,          // [1]
                  float* __restrict__ out,                  // [N,128]
                  int E) {
    __shared__ int2  s_rc[TPB][TILE];   // {row, col} interleaved -> b64 read
    __shared__ float s_w[TPB][TILE];

    const int tid       = threadIdx.x;
    const int blockBase = blockIdx.x * (TILE * TPB);

    // time = sigmoid(theta); wave-uniform scalar.
    const float th   = theta[0];
    const float tsig = 1.0f / (1.0f + __expf(-th));

    const int* ei32 = (const int*)edge_index;
    const uint32_t lds_rc0 = (uint32_t)(uintptr_t)&s_rc[0][0];
    const uint32_t lds_rc1 = (uint32_t)(uintptr_t)&s_rc[1][0];
    const uint32_t lds_w0  = (uint32_t)(uintptr_t)&s_w[0][0];
    const uint32_t lds_w1  = (uint32_t)(uintptr_t)&s_w[1][0];

    const int wave = tid >> 5;               // 8 waves per block (wave32)
    const int lane = tid & 31;               // lane*4..lane*4+3 of 128 feats
    const int EPW  = TILE / (THREADS / 32);  // 64 edges per wave
    const int wbase = wave * EPW;

    // Prologue: stage tile 0 into buffer 0.
    stage_tile(ei32, edge_weight, E, blockBase, tid, lds_rc0, lds_w0);

    #pragma unroll
    for (int t = 0; t < TPB; ++t) {
        const int tileBase = blockBase + t * TILE;
        if (tileBase >= E) break;            // block-uniform

        // Buffer (t&1) ready: own-wave async completion, then cross-wave.
        asm volatile("s_wait_asynccnt 0" ::: "memory");
        __syncthreads();

        // Overlap: kick off next tile's staging into the other buffer.
        if (t + 1 < TPB && tileBase + TILE < E) {
            stage_tile(ei32, edge_weight, E, tileBase + TILE, tid,
                       (t & 1) ? lds_rc0 : lds_rc1,
                       (t & 1) ? lds_w0  : lds_w1);
        }

        const int buf = t & 1;
        for (int i = 0; i < EPW; ++i) {
            const int li = wbase + i;
            const int e  = tileBase + li;
            if (e >= E) break;               // wave-uniform exit

            const int2  rc = s_rc[buf][li];  // broadcast ds_load_b64
            const float vv = s_w[buf][li] * tsig;

            // coalesced 512B gather of row x[col]: one b128 per lane
            const float4* xs = (const float4*)(x + (size_t)rc.y * 128);
            float4 val = xs[lane];

            const float v0 = val.x * vv;
            const float v1 = val.y * vv;
            const float v2 = val.z * vv;
            const float v3 = val.w * vv;

            float* op = out + (size_t)rc.x * 128 + lane * 4;
            // no-return fp32 hardware atomics; S_ENDPGM wait-idles at exit.
            asm volatile("global_atomic_add_f32 %0, %1, off"
                         :: "v"(op), "v"(v0) : "memory");
            asm volatile("global_atomic_add_f32 %0, %1, off offset:4"
                         :: "v"(op), "v"(v1) : "memory");
            asm volatile("global_atomic_add_f32 %0, %1, off offset:8"
                         :: "v"(op), "v"(v2) : "memory");
            asm volatile("global_atomic_add_f32 %0, %1, off offset:12"
                         :: "v"(op), "v"(v3) : "memory");
        }
    }
}

// ---------------------------------------------------------------------------
extern "C" void kernel_launch(void* const* d_in, const int* in_sizes, int n_in,
                              void* d_out, int out_size, void* d_ws, size_t ws_size,
                              hipStream_t stream) {
    const float*     x     = (const float*)d_in[0];      // [N,128] f32
    const long long* ei    = (const long long*)d_in[1];  // [2,E] i64
    const float*     ew    = (const float*)d_in[2];      // [E] f32
    const float*     theta = (const float*)d_in[3];      // [1] f32
    float*           out   = (float*)d_out;              // [N,128] f32

    const int NF = in_sizes[0];   // N * 128
    const int E  = in_sizes[2];

    const int n4 = NF / 4;
    tide_init_out<<<(n4 + 255) / 256, 256, 0, stream>>>(x, out, n4);

    const int nblocks = (E + TILE * TPB - 1) / (TILE * TPB);
    tide_edge_scatter<<<nblocks, THREADS, 0, stream>>>(x, ei, ew, theta, out, E);
}